// GSL_223338299533
// MI455X (gfx1250) — compile-verified
//
#include <hip/hip_runtime.h>

// adj:   [16, 2048, 2048] f32
// score: [16, 2048, 1]    f32
// out = adj * max(keep_row, keep_col), keep = top-1024 scores per batch.

typedef float v4f __attribute__((ext_vector_type(4)));
typedef int   v4i __attribute__((ext_vector_type(4)));

#define GSL_B 16
#define GSL_N 2048
#define GSL_K 1024   // RATE * N

#if defined(__gfx1250__)
#define HAVE_ASYNC_LDS __has_builtin(__builtin_amdgcn_global_load_async_to_lds_b128)
#else
#define HAVE_ASYNC_LDS 0
#endif

#if HAVE_ASYNC_LDS
// Builtin signature (from hipcc diagnostic): param0 is an int4* in the global
// address space (AS1, printed as "__device__"), param1 the LDS-side pointer.
typedef __attribute__((address_space(1))) v4i* gptr_b128;
typedef __attribute__((address_space(3))) v4i* lptr_b128;

// Generic->AS(1): same 64-bit representation.
__device__ __forceinline__ gptr_b128 gsl_as_g128(const void* p) {
    return (gptr_b128)(uintptr_t)p;
}
// Generic->AS(3): generic LDS pointer is {aperture_hi, lds_offset32};
// truncating to 32 bits recovers the addrspace(3) value (aperture model).
__device__ __forceinline__ lptr_b128 gsl_as_l128(const void* p) {
    return (lptr_b128)(uint32_t)(uintptr_t)p;
}

__device__ __forceinline__ void gsl_wait_async0() {
#if __has_builtin(__builtin_amdgcn_s_wait_asynccnt)
    __builtin_amdgcn_s_wait_asynccnt(0);
#else
    asm volatile("s_wait_asynccnt 0x0" ::: "memory");
#endif
}
#endif

// ---------------------------------------------------------------------------
// Kernel 1: per-batch top-k membership via rank counting (exact, stable ties).
// One block per batch. Scores staged into LDS via GLOBAL_LOAD_ASYNC_TO_LDS_B128
// (ASYNCcnt path, no VGPR round-trip); each wave waits its own async ops, then
// the workgroup barrier publishes LDS to all waves. Rank inner loop reads the
// same LDS word across all 32 lanes of a wave -> broadcast, conflict-free.
// ---------------------------------------------------------------------------
__global__ __launch_bounds__(256) void gsl_topk_keep(const float* __restrict__ score,
                                                     float* __restrict__ keep) {
    __shared__ float s[GSL_N];
    const int b = blockIdx.x;
    const int t = threadIdx.x;
    const float* __restrict__ sb = score + (size_t)b * GSL_N;

#if HAVE_ASYNC_LDS
    // 2048 floats = 512 x b128; 256 lanes x 2 issues, memory -> LDS direct.
#pragma unroll
    for (int r = 0; r < 2; ++r) {
        const int j = (t + (r << 8)) << 2;   // float index, 16B aligned
        __builtin_amdgcn_global_load_async_to_lds_b128(
            gsl_as_g128(sb + j), gsl_as_l128(&s[j]), 0, 0);
    }
    gsl_wait_async0();
#else
    for (int n = t; n < GSL_N; n += 256) s[n] = sb[n];
#endif
    __syncthreads();

    // rank(n) = #{m : s[m] > s[n]} + #{m < n : s[m] == s[n]}  (jax top_k order)
    for (int n = t; n < GSL_N; n += 256) {
        const float sn = s[n];
        int cnt = 0;
#pragma unroll 8
        for (int m = 0; m < GSL_N; ++m) {
            const float sm = s[m];
            cnt += (sm > sn) || ((sm == sn) && (m < n));
        }
        keep[(size_t)b * GSL_N + n] = (cnt < GSL_K) ? 1.0f : 0.0f;
    }
}

// ---------------------------------------------------------------------------
// Kernel 2: streaming mask-apply (the 537 MB, ~23 us bandwidth-floor kernel).
// One block per (batch, row). 256 threads x 2 x float4 = one 8 KB row.
// keep[b,i] is block-uniform -> scalar load + uniform branch:
//   kept row   : out = adj           (pure b128 NT copy, no mask reads)
//   dropped row: out = adj * keepcol (keepcol is L2-resident, 8 KB/batch)
// adj/out are single-touch and exceed the 192 MB L2 -> non-temporal hints.
// ---------------------------------------------------------------------------
__global__ __launch_bounds__(256) void gsl_mask_apply(const float* __restrict__ adj,
                                                      const float* __restrict__ keep,
                                                      float* __restrict__ out) {
    const unsigned i = blockIdx.x;          // row  0..2047
    const unsigned b = blockIdx.y;          // batch 0..15
    const size_t rowIdx  = ((size_t)b << 11) + i;      // b*N + i
    const size_t rowBase = rowIdx << 11;               // (b*N + i) * N

    const v4f* __restrict__ a = (const v4f*)(adj + rowBase);
    v4f* __restrict__ o       = (v4f*)(out + rowBase);
    const float ki = keep[rowIdx];          // uniform across block -> s_load
    const int t = threadIdx.x;

    if (ki > 0.0f) {
        // Row selected: mask == 1 everywhere, straight copy.
#pragma unroll
        for (int r = 0; r < 2; ++r) {
            const int j = t + (r << 8);
            const v4f av = __builtin_nontemporal_load(a + j);
            __builtin_nontemporal_store(av, o + j);
        }
    } else {
        // Row dropped: mask == keep[col] (0.0 or 1.0), so out = adj * keepcol.
        const v4f* __restrict__ kc = (const v4f*)(keep + ((size_t)b << 11));
#pragma unroll
        for (int r = 0; r < 2; ++r) {
            const int j = t + (r << 8);
            const v4f av = __builtin_nontemporal_load(a + j);
            const v4f kv = kc[j];           // cached (regular temporal load)
            __builtin_nontemporal_store(av * kv, o + j);
        }
    }
}

// ---------------------------------------------------------------------------
// Launcher
// ---------------------------------------------------------------------------
extern "C" void kernel_launch(void* const* d_in, const int* in_sizes, int n_in,
                              void* d_out, int out_size, void* d_ws, size_t ws_size,
                              hipStream_t stream) {
    const float* adj   = (const float*)d_in[0];   // 16*2048*2048
    const float* score = (const float*)d_in[1];   // 16*2048*1
    float* out  = (float*)d_out;
    float* keep = (float*)d_ws;                   // 16*2048 floats = 128 KB scratch

    gsl_topk_keep<<<GSL_B, 256, 0, stream>>>(score, keep);

    dim3 grid(GSL_N, GSL_B);
    gsl_mask_apply<<<grid, 256, 0, stream>>>(adj, keep, out);
}